// MyGCNConv_51565377356344
// MI455X (gfx1250) — compile-verified
//
#include <hip/hip_runtime.h>

#define D 128          // feature dim (in == out == 128)
#define LDSTR 132      // padded LDS row stride (floats) -> conflict-free fragment reads

typedef float v2f __attribute__((ext_vector_type(2)));
typedef float v8f __attribute__((ext_vector_type(8)));

// ---------------------------------------------------------------------------
// 1) init: out[i,j] = bias[j]  (bias folded into the scatter destination),
//          zero both degree arrays in ws.
// ---------------------------------------------------------------------------
__global__ void gcn_init(float* __restrict__ out, const float* __restrict__ bias,
                         float* __restrict__ degs, long total, long degTotal) {
  long i = (long)blockIdx.x * blockDim.x + threadIdx.x;
  long stride = (long)gridDim.x * blockDim.x;
  for (; i < total; i += stride) {
    out[i] = bias[i & (D - 1)];
    if (i < degTotal) degs[i] = 0.0f;
  }
}

// ---------------------------------------------------------------------------
// 2) degree histograms via hardware fp32 global atomics
// ---------------------------------------------------------------------------
__global__ void gcn_degrees(const int* __restrict__ ei,
                            float* __restrict__ degOut, float* __restrict__ degIn,
                            int nEdges) {
  int i = blockIdx.x * blockDim.x + threadIdx.x;
  if (i < nEdges) {
    unsafeAtomicAdd(&degOut[ei[i]], 1.0f);          // row = source
    unsafeAtomicAdd(&degIn[ei[nEdges + i]], 1.0f);  // col = target
  }
}

// ---------------------------------------------------------------------------
// 3) in-place deg -> deg^{-1/2} (0 stays 0)
// ---------------------------------------------------------------------------
__global__ void gcn_rsqrt(float* __restrict__ degs, int n) {
  int i = blockIdx.x * blockDim.x + threadIdx.x;
  if (i < n) {
    float d = degs[i];
    degs[i] = (d > 0.0f) ? rsqrtf(d) : 0.0f;
  }
}

// ---------------------------------------------------------------------------
// 4) y = x @ W^T using V_WMMA_F32_16X16X4_F32.
//    Block = 8 waves, one 16-row stripe; wave w -> cols [16w, 16w+16).
//    A tile (16x128) staged in LDS with padded stride.
// ---------------------------------------------------------------------------
__global__ void __launch_bounds__(256)
gcn_gemm(const float* __restrict__ x, const float* __restrict__ W,
         float* __restrict__ y) {
  __shared__ float ldsA[16 * LDSTR];

  const int t    = threadIdx.x;
  const int lane = t & 31;
  const int wave = t >> 5;
  const long rowTile = blockIdx.x;                 // 16 rows per block

  // cooperative contiguous load of the 16x128 tile (2048 floats) into LDS
  {
    const float* src = x + rowTile * (16L * D);
    int f   = t * 8;                               // 8 floats per thread, one row each
    int row = f >> 7;
    int col = f & (D - 1);
    float4 v0 = *(const float4*)(src + f);
    float4 v1 = *(const float4*)(src + f + 4);
    *(float4*)&ldsA[row * LDSTR + col]     = v0;
    *(float4*)&ldsA[row * LDSTR + col + 4] = v1;
  }
  __syncthreads();

  const int m       = lane & 15;                   // A row / B col within tile
  const int khalf   = (lane >> 4) << 1;            // lanes 16-31 hold K+2,K+3
  const int colBase = wave * 16;
  const float* arow = &ldsA[m * LDSTR];
  const float* wrow = W + (long)(colBase + m) * D; // B[k][n] = W[n][k]

  v8f acc = {};
#pragma unroll
  for (int ks = 0; ks < 32; ++ks) {                // K = 128 = 32 * 4
    int kb = ks * 4 + khalf;
    v2f a = *(const v2f*)(arow + kb);              // ds_load_b64, conflict-free
    v2f b = *(const v2f*)(wrow + kb);
    acc = __builtin_amdgcn_wmma_f32_16x16x4_f32(
        /*neg_a=*/false, a, /*neg_b=*/false, b,
        /*c_mod=*/(short)0, acc, /*reuse_a=*/false, /*reuse_b=*/false);
  }

  // C/D layout: VGPR r -> M = r + 8*(lane>=16), N = lane&15
  float* ybase = y + (rowTile * 16 + (lane >> 4) * 8) * (long)D + colBase + m;
#pragma unroll
  for (int r = 0; r < 8; ++r)
    ybase[(long)r * D] = acc[r];
}

// ---------------------------------------------------------------------------
// 5) scatter: out[row] += w_e * y[col].  One wave32 per edge; each lane
//    gathers a coalesced float4 of y[col] and issues 4 fp32 global atomics.
// ---------------------------------------------------------------------------
__global__ void __launch_bounds__(256)
gcn_scatter(const int* __restrict__ ei, const float* __restrict__ y,
            const float* __restrict__ oinv, const float* __restrict__ iinv,
            float* __restrict__ out, int nEdges) {
  int e    = (int)(((long)blockIdx.x * blockDim.x + threadIdx.x) >> 5);
  int lane = threadIdx.x & 31;
  if (e >= nEdges) return;

  int r = ei[e];
  int c = ei[nEdges + e];
  float w = oinv[r] * iinv[c];                     // > 0 for every real edge

  const float4 v = *(const float4*)(y + (long)c * D + lane * 4);
  float* dst = out + (long)r * D + lane * 4;
  unsafeAtomicAdd(dst + 0, w * v.x);
  unsafeAtomicAdd(dst + 1, w * v.y);
  unsafeAtomicAdd(dst + 2, w * v.z);
  unsafeAtomicAdd(dst + 3, w * v.w);
}

// ---------------------------------------------------------------------------
extern "C" void kernel_launch(void* const* d_in, const int* in_sizes, int n_in,
                              void* d_out, int out_size, void* d_ws, size_t ws_size,
                              hipStream_t stream) {
  const float* x    = (const float*)d_in[0];   // [N, 128]
  const int*   ei   = (const int*)d_in[1];     // [2, E]
  const float* W    = (const float*)d_in[2];   // [128, 128] row-major [out,in]
  const float* bias = (const float*)d_in[3];   // [128]
  float* out = (float*)d_out;                  // [N, 128]

  const int nNodes = in_sizes[0] / D;          // 100000
  const int nEdges = in_sizes[1] / 2;          // 1600000

  // workspace layout: [outdeg N][indeg N][y N*D]   (~52 MB)
  float* degOut = (float*)d_ws;
  float* degIn  = degOut + nNodes;
  float* y      = degIn + nNodes;

  const long total = (long)nNodes * D;

  gcn_init<<<4096, 256, 0, stream>>>(out, bias, degOut, total, 2L * nNodes);
  gcn_degrees<<<(nEdges + 255) / 256, 256, 0, stream>>>(ei, degOut, degIn, nEdges);
  gcn_rsqrt<<<(2 * nNodes + 255) / 256, 256, 0, stream>>>(degOut, 2 * nNodes);
  gcn_gemm<<<nNodes / 16, 256, 0, stream>>>(x, W, y);

  long scatterThreads = 32L * nEdges;          // one wave per edge
  gcn_scatter<<<(int)((scatterThreads + 255) / 256), 256, 0, stream>>>(
      ei, y, degOut, degIn, out, nEdges);
}